// MolDecoder_7945689497644
// MI455X (gfx1250) — compile-verified
//
#include <hip/hip_runtime.h>
#include <hip/hip_bf16.h>
#include <math.h>

typedef float v2f __attribute__((ext_vector_type(2)));
typedef float v8f __attribute__((ext_vector_type(8)));

#define N_NODES 8192
#define N_EDGES 131072
#define EMB_NF  256
#define NODE_NF 64
#define EDGE_NF 16

// ---------------------------------------------------------------------------
// Row squared-norms: one wave32 per row, shfl_xor tree reduction.
// ---------------------------------------------------------------------------
__global__ __launch_bounds__(256) void rownorm_kernel(const float* __restrict__ A,
                                                      float* __restrict__ sq) {
    const int row  = blockIdx.x * 8 + (threadIdx.x >> 5);
    const int lane = threadIdx.x & 31;
    const float* p = A + (size_t)row * EMB_NF;
    float s = 0.f;
#pragma unroll
    for (int k = lane; k < EMB_NF; k += 32) { float v = p[k]; s += v * v; }
#pragma unroll
    for (int off = 16; off > 0; off >>= 1) s += __shfl_xor(s, off, 32);
    if (lane == 0) sq[row] = s;
}

// ---------------------------------------------------------------------------
// Linear head: C[M,N] = A[M,K] @ Wt[N,K]^T + bias[N], via V_WMMA_F32_16X16X4_F32.
// One wave per 16x16 output tile. K=EMB_NF=256 -> 64 WMMA per tile.
// A operand layout (16x4 f32): lane&15 = M-row, (lane>>4)*2 = K-half, float2/lane.
// B operand (4x16): same pattern with lane&15 = N-col (B rows come from Wt rows).
// ---------------------------------------------------------------------------
template <int N>
__global__ __launch_bounds__(256) void head_gemm_kernel(const float* __restrict__ A,
                                                        const float* __restrict__ Wt,
                                                        const float* __restrict__ bias,
                                                        float* __restrict__ C) {
    const int gwave  = (int)((blockIdx.x * blockDim.x + threadIdx.x) >> 5);
    const int lane   = threadIdx.x & 31;
    const int ntiles = N / 16;
    const int ti = gwave / ntiles;          // M tile index
    const int tj = gwave % ntiles;          // N tile index
    const int lrow  = lane & 15;
    const int khalf = (lane >> 4) << 1;     // 0 or 2

    const float* ap = A  + (size_t)(ti * 16 + lrow) * EMB_NF + khalf;
    const float* bp = Wt + (size_t)(tj * 16 + lrow) * EMB_NF + khalf;

    v8f acc = {0.f, 0.f, 0.f, 0.f, 0.f, 0.f, 0.f, 0.f};
#pragma unroll 8
    for (int k = 0; k < EMB_NF; k += 4) {
        v2f a = *(const v2f*)(ap + k);
        v2f b = *(const v2f*)(bp + k);
        acc = __builtin_amdgcn_wmma_f32_16x16x4_f32(false, a, false, b,
                                                    (short)0, acc, false, false);
    }

    const int col   = tj * 16 + lrow;
    const float bv  = bias[col];
    const int rbase = ti * 16 + ((lane >> 4) << 3);   // lanes 16-31 own rows +8
#pragma unroll
    for (int r = 0; r < 8; ++r)
        C[(size_t)(rbase + r) * N + col] = acc[r] + bv;
}

// ---------------------------------------------------------------------------
// adj_pred: G = X @ X^T (Gram), d2 = sq_i + sq_j - 2G, sigmoid, zero diagonal.
// Block = 256 thr = 8 waves arranged 4(M) x 2(N); each wave owns a 32x64
// register tile (2x4 accumulator tiles of 16x16). Per K=4 step: 6 b64 loads
// feed 8 WMMAs (A reused across 4 N-tiles, B across 2 M-tiles).
// Block tile = 128x128; grid = 64x64 (exact), EXEC all-ones everywhere.
// ---------------------------------------------------------------------------
__global__ __launch_bounds__(256) void adj_kernel(const float* __restrict__ X,
                                                  const float* __restrict__ sq,
                                                  const float* __restrict__ Wp,
                                                  const float* __restrict__ bp,
                                                  float* __restrict__ out) {
    const int lane = threadIdx.x & 31;
    const int wid  = threadIdx.x >> 5;      // 0..7
    const int wi = wid >> 1;                // 0..3 -> M
    const int wj = wid & 1;                 // 0..1 -> N
    const int i0 = blockIdx.y * 128 + wi * 32;
    const int j0 = blockIdx.x * 128 + wj * 64;

    const int lrow  = lane & 15;
    const int khalf = (lane >> 4) << 1;

    const float* arow[2];
    const float* brow[4];
#pragma unroll
    for (int m = 0; m < 2; ++m)
        arow[m] = X + (size_t)(i0 + m * 16 + lrow) * EMB_NF + khalf;
#pragma unroll
    for (int n = 0; n < 4; ++n)
        brow[n] = X + (size_t)(j0 + n * 16 + lrow) * EMB_NF + khalf;

    const v8f zero = {0.f, 0.f, 0.f, 0.f, 0.f, 0.f, 0.f, 0.f};
    v8f acc[2][4];
#pragma unroll
    for (int m = 0; m < 2; ++m)
#pragma unroll
        for (int n = 0; n < 4; ++n) acc[m][n] = zero;

#pragma unroll 4
    for (int k = 0; k < EMB_NF; k += 4) {
        v2f a[2], b[4];
#pragma unroll
        for (int m = 0; m < 2; ++m) a[m] = *(const v2f*)(arow[m] + k);
#pragma unroll
        for (int n = 0; n < 4; ++n) b[n] = *(const v2f*)(brow[n] + k);
#pragma unroll
        for (int m = 0; m < 2; ++m)
#pragma unroll
            for (int n = 0; n < 4; ++n)
                acc[m][n] = __builtin_amdgcn_wmma_f32_16x16x4_f32(
                    false, a[m], false, b[n], (short)0, acc[m][n], false, false);
    }

    const float Wc = Wp[0];
    const float bc = bp[0];
#pragma unroll
    for (int m = 0; m < 2; ++m) {
        const int rbase = i0 + m * 16 + ((lane >> 4) << 3);
#pragma unroll
        for (int n = 0; n < 4; ++n) {
            const int col   = j0 + n * 16 + lrow;
            const float sqj = sq[col];
#pragma unroll
            for (int r = 0; r < 8; ++r) {
                const int row = rbase + r;
                const float d2 = sq[row] + sqj - 2.0f * acc[m][n][r];
                const float z  = Wc * d2 + bc;
                float s = 1.0f / (1.0f + __expf(-z));
                if (row == col) s = 0.0f;            // remove self-loops
                out[(size_t)row * N_NODES + col] = s;
            }
        }
    }
}

// ---------------------------------------------------------------------------
// Launch: inputs are (node_emb, edge_emb, node_w, node_b, edge_w, edge_b, W, b).
// d_out = [recon_node | recon_edge | adj_pred] flat f32.
// ---------------------------------------------------------------------------
extern "C" void kernel_launch(void* const* d_in, const int* in_sizes, int n_in,
                              void* d_out, int out_size, void* d_ws, size_t ws_size,
                              hipStream_t stream) {
    const float* node_emb = (const float*)d_in[0];
    const float* edge_emb = (const float*)d_in[1];
    const float* node_w   = (const float*)d_in[2];
    const float* node_b   = (const float*)d_in[3];
    const float* edge_w   = (const float*)d_in[4];
    const float* edge_b   = (const float*)d_in[5];
    const float* Wp       = (const float*)d_in[6];
    const float* bp       = (const float*)d_in[7];

    float* out_node = (float*)d_out;
    float* out_edge = out_node + (size_t)N_NODES * NODE_NF;
    float* out_adj  = out_edge + (size_t)N_EDGES * EDGE_NF;
    float* sq       = (float*)d_ws;   // 8192 floats = 32 KB scratch

    // 1) row squared-norms (needed by adj epilogue)
    rownorm_kernel<<<N_NODES / 8, 256, 0, stream>>>(node_emb, sq);

    // 2) node head: 512 M-tiles x 4 N-tiles = 2048 waves -> 256 blocks
    head_gemm_kernel<NODE_NF>
        <<<(N_NODES / 16) * (NODE_NF / 16) / 8, 256, 0, stream>>>(
            node_emb, node_w, node_b, out_node);

    // 3) edge head: 8192 M-tiles x 1 N-tile = 8192 waves -> 1024 blocks
    head_gemm_kernel<EDGE_NF>
        <<<(N_EDGES / 16) * (EDGE_NF / 16) / 8, 256, 0, stream>>>(
            edge_emb, edge_w, edge_b, out_edge);

    // 4) adjacency: 128x128 block tiles, exact 64x64 grid
    dim3 grid(N_NODES / 128, N_NODES / 128);
    adj_kernel<<<grid, 256, 0, stream>>>(node_emb, sq, Wp, bp, out_adj);
}